// DBSCAN_cluster_11699490914464
// MI455X (gfx1250) — compile-verified
//
#include <hip/hip_runtime.h>
#include <stdint.h>

// ---------------------------------------------------------------------------
// DBSCAN (eps=1.5 on integer grid == 8-neighborhood), H=1024, W=2048.
//   core      = active & (3x3 active count >= 4)
//   labels    = connected components of core (8-conn), label = min index
//   border    = active non-core joins min neighboring core root
//   filter    size >= 30, rank clusters by mean column, one-hot 26 channels
//
// MI455X notes: zero-FLOP, bandwidth bound (218 MB output -> ~9.4us floor at
// 23.3 TB/s). No matrix math -> no WMMA; the CDNA5 paths used are
// global_load_async_to_lds_b32 + s_wait_asynccnt (async staging of the halo
// tile) and ds_add_u32/u64 LDS atomics for per-block cluster stats.
// ---------------------------------------------------------------------------

#define HH 1024
#define WW 2048
#define HW (HH * WW)
#define MIN_SAMPLES 4
#define MIN_CLUSTER 30
#define N_SLOTS 26
#define MAXK (HW / MIN_CLUSTER + 1)   // max possible valid clusters

typedef float  vf4 __attribute__((ext_vector_type(4)));
typedef int    vi4 __attribute__((ext_vector_type(4)));

// ------------------------- union-find helpers ------------------------------
__device__ __forceinline__ int findRootRO(const int* __restrict__ P, int x) {
    int p = P[x];
    while (p != x) { x = p; p = P[x]; }
    return x;
}
__device__ __forceinline__ int findRootPH(int* P, int x) {  // path-halving
    int p = P[x];
    while (p != x) {
        int gp = P[p];
        if (gp != p) P[x] = gp;   // benign race: monotone pointer shortening
        x = p; p = gp;
    }
    return x;
}
// Lock-free union, always attach larger root under smaller root.
// => final root of every component is the component's MIN index (matches
//    the reference's converged min-pool labels exactly).
__device__ __forceinline__ void unite(int* P, int a, int b) {
    a = findRootPH(P, a);
    b = findRootPH(P, b);
    while (a != b) {
        if (a < b) { int t = a; a = b; b = t; }     // ensure a > b
        int old = atomicCAS(&P[a], a, b);
        if (old == a) return;
        a = findRootPH(P, old);
        b = findRootPH(P, b);
    }
}

// ------------------- K0: stencil (core mask) + array init ------------------
#define TW 32
#define THT 8
#define TS (TW + 4)   // LDS row stride (padded)

__global__ __launch_bounds__(256) void k0_core_init(
        const float* __restrict__ mask, int* __restrict__ P,
        unsigned char* __restrict__ flags, int* __restrict__ sizes,
        unsigned long long* __restrict__ colsum, int* __restrict__ chan,
        int* __restrict__ counter)
{
    __shared__ float tile[(THT + 2) * TS];

    const int t  = threadIdx.x;
    const int tx = t & (TW - 1);
    const int ty = t >> 5;
    const int bx = blockIdx.x * TW;
    const int by = blockIdx.y * THT;

    // Deterministic per-call init of label-indexed arrays (grid == HW threads)
    const int gid = (int)((blockIdx.y * gridDim.x + blockIdx.x) * 256u + (unsigned)t);
    sizes[gid]  = 0;
    colsum[gid] = 0ull;
    chan[gid]   = N_SLOTS;
    if (gid == 0) *counter = 0;

    // Async-stage the (THT+2)x(TW+2) halo tile straight into LDS (CDNA5 async
    // path: no VGPR round trip, tracked by ASYNCcnt).
    for (int i = t; i < (THT + 2) * (TW + 2); i += 256) {
        const int ly = i / (TW + 2);
        const int lx = i - ly * (TW + 2);
        int gy = by + ly - 1; gy = gy < 0 ? 0 : (gy >= HH ? HH - 1 : gy);
        int gx = bx + lx - 1; gx = gx < 0 ? 0 : (gx >= WW ? WW - 1 : gx);
        const float* gp = mask + (size_t)gy * WW + gx;
        unsigned lp = (unsigned)(uintptr_t)(&tile[ly * TS + lx]);  // LDS byte offset
        asm volatile("global_load_async_to_lds_b32 %0, %1, off"
                     :: "v"(lp), "v"(gp) : "memory");
    }
    asm volatile("s_wait_asynccnt 0x0" ::: "memory");   // this wave's DMAs done
    __syncthreads();                                    // all waves' DMAs done

    const int x = bx + tx, y = by + ty;
    const bool act = tile[(ty + 1) * TS + (tx + 1)] > 0.1f;
    int cnt = 0;
#pragma unroll
    for (int dy = -1; dy <= 1; ++dy)
#pragma unroll
        for (int dx = -1; dx <= 1; ++dx) {
            const int yy = y + dy, xx = x + dx;
            if (yy >= 0 && yy < HH && xx >= 0 && xx < WW)
                cnt += (tile[(ty + 1 + dy) * TS + (tx + 1 + dx)] > 0.1f) ? 1 : 0;
        }
    const bool core = act && (cnt >= MIN_SAMPLES);
    const int p = y * WW + x;
    flags[p] = (unsigned char)((act ? 1u : 0u) | (core ? 2u : 0u));
    P[p] = p;
}

// ---------------- K1: single-pass union-find merge (8-conn) -----------------
__global__ __launch_bounds__(256) void k1_merge(
        const unsigned char* __restrict__ flags, int* __restrict__ P)
{
    const int p = blockIdx.x * 256 + threadIdx.x;
    if (p >= HW) return;
    if (!(flags[p] & 2u)) return;
    const int y = p >> 11, x = p & (WW - 1);
    if (x > 0 && (flags[p - 1] & 2u)) unite(P, p, p - 1);
    if (y > 0) {
        const int q = p - WW;
        if (x > 0      && (flags[q - 1] & 2u)) unite(P, p, q - 1);
        if (             (flags[q]     & 2u)) unite(P, p, q);
        if (x < WW - 1 && (flags[q + 1] & 2u)) unite(P, p, q + 1);
    }
}

// -------- K2: flatten core labels + attach border pixels (min root) ---------
__global__ __launch_bounds__(256) void k2_final(
        const unsigned char* __restrict__ flags, const int* __restrict__ P,
        int* __restrict__ F)
{
    const int p = blockIdx.x * 256 + threadIdx.x;
    if (p >= HW) return;
    const unsigned f = flags[p];
    int out = HW;                                    // BIG sentinel
    if (f & 2u) {
        out = findRootRO(P, p);
    } else if (f & 1u) {
        const int y = p >> 11, x = p & (WW - 1);
        int m = HW;
#pragma unroll
        for (int dy = -1; dy <= 1; ++dy)
#pragma unroll
            for (int dx = -1; dx <= 1; ++dx) {
                if (dy == 0 && dx == 0) continue;
                const int yy = y + dy, xx = x + dx;
                if (yy < 0 || yy >= HH || xx < 0 || xx >= WW) continue;
                const int q = yy * WW + xx;
                if (flags[q] & 2u) {
                    const int r = findRootRO(P, q);
                    m = m < r ? m : r;
                }
            }
        out = m;                                     // stays HW -> noise
    }
    F[p] = out;
}

// -------- K3: per-cluster size + exact integer column sum (LDS hash) --------
#define HASHSZ 512
__global__ __launch_bounds__(256) void k3_stats(
        const int* __restrict__ F, int* __restrict__ sizes,
        unsigned long long* __restrict__ colsum)
{
    __shared__ int hkey[HASHSZ];
    __shared__ int hcnt[HASHSZ];
    __shared__ unsigned long long hsum[HASHSZ];
    for (int i = threadIdx.x; i < HASHSZ; i += 256) {
        hkey[i] = -1; hcnt[i] = 0; hsum[i] = 0ull;
    }
    __syncthreads();

    const int p = blockIdx.x * 256 + threadIdx.x;
    const int l = (p < HW) ? F[p] : HW;
    if (l < HW) {
        const unsigned long long col = (unsigned long long)(p & (WW - 1));
        unsigned h = ((unsigned)l * 2654435761u) & (HASHSZ - 1);
        for (;;) {                                   // <=256 keys in 512 slots
            const int k = atomicCAS(&hkey[h], -1, l);
            if (k == -1 || k == l) {
                atomicAdd(&hcnt[h], 1);              // ds_add_u32
                atomicAdd(&hsum[h], col);            // ds_add_u64
                break;
            }
            h = (h + 1) & (HASHSZ - 1);
        }
    }
    __syncthreads();
    for (int i = threadIdx.x; i < HASHSZ; i += 256) {
        const int k = hkey[i];
        if (k >= 0) {
            atomicAdd(&sizes[k], hcnt[i]);
            atomicAdd(&colsum[k], hsum[i]);
        }
    }
}

// ----------------- K4: compact valid clusters (size >= 30) ------------------
__global__ __launch_bounds__(256) void k4_collect(
        const int* __restrict__ sizes, const unsigned long long* __restrict__ colsum,
        int* __restrict__ counter, int* __restrict__ labs,
        int* __restrict__ szs, unsigned long long* __restrict__ csums)
{
    const int i = blockIdx.x * 256 + threadIdx.x;
    if (i >= HW) return;
    const int s = sizes[i];
    if (s >= MIN_CLUSTER) {
        const int j = atomicAdd(counter, 1);         // order nondet, result set det
        labs[j] = i; szs[j] = s; csums[j] = colsum[i];
    }
}

// --- K5: rank by mean column (exact rational compare, label tie-break) ------
__global__ __launch_bounds__(256) void k5_rank(
        const int* __restrict__ counter, const int* __restrict__ labs,
        const int* __restrict__ szs, const unsigned long long* __restrict__ csums,
        int* __restrict__ chan)
{
    const int j = blockIdx.x * 256 + threadIdx.x;
    const int K = *counter;
    if (j >= K) return;
    const unsigned long long cj = csums[j];
    const unsigned long long sj = (unsigned long long)szs[j];
    const int lj = labs[j];
    int r = 0;
    for (int i = 0; i < K; ++i) {
        const unsigned long long a = csums[i] * sj;   // mean_i vs mean_j, exact:
        const unsigned long long b = cj * (unsigned long long)szs[i];
        r += (a < b) || (a == b && labs[i] < lj);     // <= 2^54, fits u64
    }
    if (r < N_SLOTS) chan[lj] = r;                    // else stays N_SLOTS
}

// ---- K6: 26-channel one-hot expansion, 4 pixels/thread, b128 NT stores -----
__global__ __launch_bounds__(256) void k6_out(
        const int* __restrict__ F, const int* __restrict__ chan,
        float* __restrict__ out)
{
    const int p4 = blockIdx.x * 256 + threadIdx.x;    // quad index
    if (p4 >= HW / 4) return;
    const int p = p4 * 4;

    const vi4 l4 = *(const vi4*)&F[p];                // global_load_b128
    int c0 = (l4.x < HW) ? chan[l4.x] : N_SLOTS;
    int c1 = (l4.y < HW) ? chan[l4.y] : N_SLOTS;
    int c2 = (l4.z < HW) ? chan[l4.z] : N_SLOTS;
    int c3 = (l4.w < HW) ? chan[l4.w] : N_SLOTS;

#pragma unroll
    for (int k = 0; k < N_SLOTS; ++k) {
        vf4 v;
        v.x = (k == c0) ? 1.0f : 0.0f;
        v.y = (k == c1) ? 1.0f : 0.0f;
        v.z = (k == c2) ? 1.0f : 0.0f;
        v.w = (k == c3) ? 1.0f : 0.0f;
        __builtin_nontemporal_store(v, (vf4*)&out[(size_t)k * HW + p]);
    }
}

// ---------------------------------------------------------------------------
extern "C" void kernel_launch(void* const* d_in, const int* in_sizes, int n_in,
                              void* d_out, int out_size, void* d_ws, size_t ws_size,
                              hipStream_t stream)
{
    const float* mask = (const float*)d_in[0];
    float* out = (float*)d_out;
    char* ws = (char*)d_ws;

    // Workspace layout (~53 MB total)
    int* P       = (int*)(ws + 0);                       //  8 MiB
    int* F       = (int*)(ws + (8u  << 20));             //  8 MiB
    int* sizes   = (int*)(ws + (16u << 20));             //  8 MiB
    unsigned long long* colsum = (unsigned long long*)(ws + (24u << 20)); // 16 MiB
    int* chan    = (int*)(ws + (40u << 20));             //  8 MiB
    unsigned char* flags = (unsigned char*)(ws + (48u << 20));            //  2 MiB
    int* labs    = (int*)(ws + (50u << 20));             //  <=0.3 MiB
    int* szs     = (int*)(ws + (50u << 20) + (512u << 10));
    unsigned long long* csums = (unsigned long long*)(ws + (51u << 20));  //  <=0.6 MiB
    int* counter = (int*)(ws + (52u << 20));

    const dim3 g0(WW / TW, HH / THT);                    // 64 x 128 = 8192 blocks
    const int NB = HW / 256;                             // 8192
    const int RB = (MAXK + 255) / 256;

    k0_core_init<<<g0, 256, 0, stream>>>(mask, P, flags, sizes, colsum, chan, counter);
    k1_merge   <<<NB, 256, 0, stream>>>(flags, P);
    k2_final   <<<NB, 256, 0, stream>>>(flags, P, F);
    k3_stats   <<<NB, 256, 0, stream>>>(F, sizes, colsum);
    k4_collect <<<NB, 256, 0, stream>>>(sizes, colsum, counter, labs, szs, csums);
    k5_rank    <<<RB, 256, 0, stream>>>(counter, labs, szs, csums, chan);
    k6_out     <<<HW / 4 / 256, 256, 0, stream>>>(F, chan, out);
}